// SAGPool_39848706572350
// MI455X (gfx1250) — compile-verified
//
#include <hip/hip_runtime.h>
#include <hip/hip_bf16.h>

// ---------------- problem constants (match reference) ----------------
#define BGRAPH 256
#define NNODE  512
#define DFEAT  256
#define NDEG   32
#define NEDGE  (BGRAPH * NNODE * NDEG)   // 4,194,304
#define KKEEP  409
#define NTOT   (BGRAPH * NNODE)          // 131,072
#define XOUT_ELEMS (BGRAPH * KKEEP * DFEAT)  // 26,804,224

typedef __attribute__((ext_vector_type(2))) float v2f;
typedef __attribute__((ext_vector_type(8))) float v8f;

// ---------------------------------------------------------------------
// K1: per-node score GEMV h = x @ W via V_WMMA_F32_16X16X4_F32 (full fp32).
// A (16x4) = W in row M=0 only; B (4x16) = transposed 16-node x-tile.
// ISA layouts (CDNA5 ISA):
//   A: lane L holds M=L%16; VGPR j -> K = (L<16?0:2)+j
//   B: lane L holds N=L%16; VGPR j -> K = (L<16?0:2)+j
//   C/D: VGPR0 lanes 0..15 -> (M=0, N=lane)  => scores for 16 nodes.
// Both per-lane operand pairs are K-contiguous -> aligned float2 loads.
// Also initializes deg[] = 1.0f (self-loop).
// ---------------------------------------------------------------------
__global__ __launch_bounds__(256)
void k_score_gemv_wmma(const float* __restrict__ x, const float* __restrict__ W,
                       float* __restrict__ h, float* __restrict__ deg) {
    const int lane = threadIdx.x & 31;
    const int wave = threadIdx.x >> 5;
    const int nodeBase = (blockIdx.x * 8 + wave) * 16;
    const int n = lane & 15;                  // node (N) / row (M) within tile
    const int koff = (lane < 16) ? 0 : 2;     // K offset for this half-wave
    const float am = (n == 0) ? 1.0f : 0.0f;  // only A row M=0 is live (holds W)

    const float* xrow = x + (size_t)(nodeBase + n) * DFEAT;

    v8f c = {};
    #pragma unroll 4
    for (int kb = 0; kb < DFEAT; kb += 4) {
        const float2 bx = *(const float2*)(xrow + kb + koff);
        const float2 wx = *(const float2*)(W + kb + koff);
        v2f bvec; bvec.x = bx.x;      bvec.y = bx.y;
        v2f avec; avec.x = wx.x * am; avec.y = wx.y * am;
        // 8 args: (neg_a, A, neg_b, B, c_mod, C, reuse_a, reuse_b)
        c = __builtin_amdgcn_wmma_f32_16x16x4_f32(
                false, avec, false, bvec, (short)0, c, false, false);
    }
    if (lane < 16) {
        h[nodeBase + lane]   = c[0];   // D[M=0][N=lane]
        deg[nodeBase + lane] = 1.0f;   // self-loop
    }
}

// K2: degree accumulation over edge destinations (adds of 1.0f are exact).
__global__ __launch_bounds__(256)
void k_degree(const int* __restrict__ col, float* __restrict__ deg) {
    const int e = blockIdx.x * blockDim.x + threadIdx.x;
    if (e < NEDGE) atomicAdd(&deg[col[e]], 1.0f);
}

// K3a: dinv = rsqrt(deg); score = self-loop term h/deg + bias.
__global__ __launch_bounds__(256)
void k_selfloop(const float* __restrict__ h, const float* __restrict__ deg,
                const float* __restrict__ b, float* __restrict__ dinv,
                float* __restrict__ score) {
    const int i = blockIdx.x * blockDim.x + threadIdx.x;
    if (i >= NTOT) return;
    const float dg = deg[i];
    const float di = rsqrtf(dg);
    dinv[i]  = di;
    score[i] = h[i] / dg + b[0];
}

// K3b: score[col] += dinv[row]*dinv[col]*h[row] over all edges.
__global__ __launch_bounds__(256)
void k_edge_accum(const int* __restrict__ row, const int* __restrict__ col,
                  const float* __restrict__ h, const float* __restrict__ dinv,
                  float* __restrict__ score) {
    const int e = blockIdx.x * blockDim.x + threadIdx.x;
    if (e >= NEDGE) return;
    const int r = row[e], c = col[e];
    atomicAdd(&score[c], dinv[r] * dinv[c] * h[r]);
}

// K4: per-graph stable top-K + relabel. One block (512 threads) per graph.
// rank matches jax.lax.top_k stability; kept nodes in ascending node order.
// Since every graph keeps exactly K nodes, newidx == cumsum(mask)-1 == g*K+local.
__global__ __launch_bounds__(512)
void k_topk(const float* __restrict__ score,
            int* __restrict__ keptflag, int* __restrict__ newidx,
            int* __restrict__ perm_nodes, float* __restrict__ tanhsc,
            float* __restrict__ out_batch, float* __restrict__ out_perm) {
    __shared__ float s[NNODE];
    __shared__ int   kf[NNODE];
    const int g = blockIdx.x;
    const int i = threadIdx.x;
    const float si = score[g * NNODE + i];
    s[i] = si;
    __syncthreads();

    int rank = 0;
    #pragma unroll 8
    for (int j = 0; j < NNODE; ++j) {
        const float sj = s[j];
        rank += (sj > si) || (sj == si && j < i);
    }
    const int keep = (rank < KKEEP) ? 1 : 0;
    kf[i] = keep;
    keptflag[g * NNODE + i] = keep;
    __syncthreads();

    int local = 0;
    for (int j = 0; j < i; ++j) local += kf[j];

    if (keep) {
        const int p = g * KKEEP + local;
        perm_nodes[p] = g * NNODE + i;
        tanhsc[p]     = tanhf(si);
        newidx[g * NNODE + i] = p;
        out_batch[p] = (float)g;
        out_perm[p]  = (float)(g * NNODE + i);
    } else {
        newidx[g * NNODE + i] = -1;
    }
}

// K5: x_out[j] = x[perm[j]] * tanh(score[perm[j]]), float4-vectorized gather.
__global__ __launch_bounds__(256)
void k_gather(const float* __restrict__ x, const int* __restrict__ perm_nodes,
              const float* __restrict__ tanhsc, float* __restrict__ xout) {
    const int t = blockIdx.x * blockDim.x + threadIdx.x;
    const int TOT = BGRAPH * KKEEP * (DFEAT / 4);
    if (t >= TOT) return;
    const int j  = t >> 6;          // row (D/4 == 64)
    const int d4 = t & 63;
    const int node = perm_nodes[j];
    const float tt = tanhsc[j];
    float4 v = ((const float4*)x)[(size_t)node * (DFEAT / 4) + d4];
    v.x *= tt; v.y *= tt; v.z *= tt; v.w *= tt;
    ((float4*)xout)[(size_t)j * (DFEAT / 4) + d4] = v;
}

// K6: edge mask + relabeled edge index (masked -> -1), written as float.
__global__ __launch_bounds__(256)
void k_edges_out(const int* __restrict__ row, const int* __restrict__ col,
                 const int* __restrict__ keptflag, const int* __restrict__ newidx,
                 float* __restrict__ out_row, float* __restrict__ out_col,
                 float* __restrict__ out_mask) {
    const int e = blockIdx.x * blockDim.x + threadIdx.x;
    if (e >= NEDGE) return;
    const int r = row[e], c = col[e];
    const int m = keptflag[r] & keptflag[c];
    out_row[e]  = m ? (float)newidx[r] : -1.0f;
    out_col[e]  = m ? (float)newidx[c] : -1.0f;
    out_mask[e] = (float)m;
}

extern "C" void kernel_launch(void* const* d_in, const int* in_sizes, int n_in,
                              void* d_out, int out_size, void* d_ws, size_t ws_size,
                              hipStream_t stream) {
    const float* x  = (const float*)d_in[0];
    const int*   ei = (const int*)d_in[1];       // [2, E]
    // d_in[2] = batch (unused; batch_out is derivable)
    const float* W  = (const float*)d_in[3];
    const float* b  = (const float*)d_in[4];

    const int* row = ei;
    const int* col = ei + NEDGE;

    // workspace layout (floats): h, deg, dinv, score, keptflag, newidx, perm, tanh
    float* ws = (float*)d_ws;
    float* h        = ws;
    float* deg      = ws + NTOT;
    float* dinv     = ws + 2 * NTOT;
    float* score    = ws + 3 * NTOT;
    int*   keptflag = (int*)(ws + 4 * NTOT);
    int*   newidx   = (int*)(ws + 5 * NTOT);
    int*   permn    = (int*)(ws + 6 * NTOT);
    float* tanhsc   = ws + 6 * NTOT + BGRAPH * KKEEP;

    // output layout (floats, reference return order)
    float* out      = (float*)d_out;
    float* xout     = out;
    float* out_row  = out + XOUT_ELEMS;
    float* out_col  = out_row + NEDGE;
    float* out_mask = out + XOUT_ELEMS + 2 * NEDGE;
    float* out_bat  = out + XOUT_ELEMS + 3 * NEDGE;
    float* out_perm = out_bat + BGRAPH * KKEEP;

    const int edgeBlocks = NEDGE / 256;                     // 16384
    const int nodeBlocks = NTOT / 256;                      // 512
    const int gathBlocks = (BGRAPH * KKEEP * (DFEAT / 4) + 255) / 256;

    k_score_gemv_wmma<<<NTOT / 128, 256, 0, stream>>>(x, W, h, deg);
    k_degree<<<edgeBlocks, 256, 0, stream>>>(col, deg);
    k_selfloop<<<nodeBlocks, 256, 0, stream>>>(h, deg, b, dinv, score);
    k_edge_accum<<<edgeBlocks, 256, 0, stream>>>(row, col, h, dinv, score);
    k_topk<<<BGRAPH, 512, 0, stream>>>(score, keptflag, newidx, permn, tanhsc,
                                       out_bat, out_perm);
    k_gather<<<gathBlocks, 256, 0, stream>>>(x, permn, tanhsc, xout);
    k_edges_out<<<edgeBlocks, 256, 0, stream>>>(row, col, keptflag, newidx,
                                                out_row, out_col, out_mask);
}